// BiPairwiseNegativeCELoss_4776003633259
// MI455X (gfx1250) — compile-verified
//
#include <hip/hip_runtime.h>
#include <math.h>

typedef float v2f __attribute__((ext_vector_type(2)));
typedef float v8f __attribute__((ext_vector_type(8)));

#define B_ROWS 16384
#define DIM    128
#define BIGF   1.0e6f

#define BM 64          // rows per block (4 waves x 16)
#define BNO 64         // columns staged per outer iteration (== BM, aligned)
#define LDS_STRIDE 132 // 128 + 4 pad -> conflict-free 8B reads, 16B-aligned rows

__device__ __forceinline__ float softplus_f(float x) {
    // numerically stable softplus
    return fmaxf(x, 0.0f) + log1pf(expf(-fabsf(x)));
}

__device__ __forceinline__ v8f wmma_f32(v2f a, v2f b, v8f c) {
    return __builtin_amdgcn_wmma_f32_16x16x4_f32(
        false, a, false, b, (short)0, c, false, false);
}

__device__ __forceinline__ unsigned lds_offset_of(const void* p) {
    // generic LDS pointer: addr[31:0] is the offset within the wave's LDS
    return (unsigned)(unsigned long long)p;
}

// Issue one per-lane async global->LDS 16B copy (tracked by ASYNCcnt).
__device__ __forceinline__ void async_copy_b128(unsigned lds_off, const float* gaddr) {
    asm volatile("global_load_async_to_lds_b128 %0, %1, off"
                 :: "v"(lds_off), "v"(gaddr) : "memory");
}

__device__ __forceinline__ void wait_asynccnt0() {
    asm volatile("s_wait_asynccnt 0x0" ::: "memory");
}

// ---------------- Kernel 1: pairwise term ----------------
// one wave per row; lane loads float4 (coalesced 512B per wave)
__global__ __launch_bounds__(256) void pairwise_kernel(
        const float* __restrict__ q, const float* __restrict__ d,
        const float* __restrict__ nd, float* __restrict__ ws) {
    int wave = threadIdx.x >> 5;
    int lane = threadIdx.x & 31;
    int row  = blockIdx.x * 8 + wave;
    const float4 qv = *(const float4*)(q  + row * DIM + lane * 4);
    const float4 dv = *(const float4*)(d  + row * DIM + lane * 4);
    const float4 nv = *(const float4*)(nd + row * DIM + lane * 4);
    float p = qv.x * dv.x + qv.y * dv.y + qv.z * dv.z + qv.w * dv.w;
    float n = qv.x * nv.x + qv.y * nv.y + qv.z * nv.z + qv.w * nv.w;
    #pragma unroll
    for (int m = 16; m >= 1; m >>= 1) {
        p += __shfl_xor(p, m, 32);
        n += __shfl_xor(n, m, 32);
    }
    __shared__ float red[8];
    if (lane == 0) red[wave] = softplus_f(n - p);
    __syncthreads();
    if (threadIdx.x == 0) {
        float s = 0.f;
        #pragma unroll
        for (int i = 0; i < 8; ++i) s += red[i];
        atomicAdd(&ws[0], s);
    }
}

// ---------------- Kernel 2: in-batch GEMM + masked row-max ----------------
// Each block: 64 rows of Q resident in registers as A-fragments.
// 64-column D panels double-buffered in LDS via async global->LDS copies,
// consumed as four 16-col sub-tiles with register-resident B fragments and
// four interleaved independent WMMA accumulation chains.
__global__ __launch_bounds__(128) void gemm_rowmax_kernel(
        const float* __restrict__ q, const float* __restrict__ d,
        float* __restrict__ ws) {
    __shared__ __align__(16) float Bs[2][BNO * LDS_STRIDE];

    const int tid  = threadIdx.x;
    const int wave = tid >> 5;
    const int lane = tid & 31;
    const int half = lane >> 4;   // selects K-pair within fragment
    const int lm   = lane & 15;   // M index for A, N index for B
    const int koff = half * 2;
    const int row_base = blockIdx.x * BM + wave * 16;
    const int diag_n0  = blockIdx.x * BM;  // only panel that can hold the diagonal

    // Per-thread staging coordinates: 2048 float4 per panel, 16 per thread
    // f = tid + 128*i -> row = f>>5 (0..63), col4 = f&31
    unsigned lds_base[2];
    lds_base[0] = lds_offset_of(&Bs[0][0]);
    lds_base[1] = lds_offset_of(&Bs[1][0]);

    // Issue async panel load: D rows [n0, n0+64) x 128 floats -> Bs[buf]
    auto stage_async = [&](int buf, int n0) {
        #pragma unroll
        for (int i = 0; i < 16; ++i) {
            int f  = tid + 128 * i;
            int rr = f >> 5;
            int c4 = f & 31;
            unsigned loff = lds_base[buf] + (unsigned)((rr * LDS_STRIDE + c4 * 4) * 4);
            async_copy_b128(loff, d + (n0 + rr) * DIM + c4 * 4);
        }
    };

    // Preload A fragments for this wave's 16 rows: 32 k-steps x v2f = 64 VGPRs
    v2f a[32];
    const float* qrow = q + (row_base + lm) * DIM + koff;
    #pragma unroll
    for (int kk = 0; kk < 32; ++kk)
        a[kk] = *(const v2f*)(qrow + kk * 4);

    float rmax[8], diag[8];
    #pragma unroll
    for (int r = 0; r < 8; ++r) { rmax[r] = -3.0e38f; diag[r] = 0.f; }

    // Prologue: fill buffer 0
    stage_async(0, 0);
    wait_asynccnt0();
    __syncthreads();

    for (int n0 = 0; n0 < B_ROWS; n0 += BNO) {
        const int cur = (n0 >> 6) & 1;
        // Kick off next panel into the other buffer; it overlaps the
        // 128 WMMAs below and is waited on at the end of this iteration.
        if (n0 + BNO < B_ROWS)
            stage_async(cur ^ 1, n0 + BNO);

        const bool diag_panel = (n0 == diag_n0);  // uniform scalar branch

        #pragma unroll
        for (int sub = 0; sub < 4; ++sub) {
            // Preload entire B tile into registers: one dscnt wait, then
            // back-to-back WMMAs.
            v2f b[32];
            const float* brow = &Bs[cur][(sub * 16 + lm) * LDS_STRIDE + koff];
            #pragma unroll
            for (int kk = 0; kk < 32; ++kk)
                b[kk] = *(const v2f*)(brow + kk * 4);

            // Four independent accumulation chains (K split), interleaved
            // for matrix-pipe ILP; merged exactly at the end.
            v8f c0 = {}, c1 = {}, c2 = {}, c3 = {};
            #pragma unroll
            for (int kk = 0; kk < 8; ++kk) {
                c0 = wmma_f32(a[kk],      b[kk],      c0);
                c1 = wmma_f32(a[kk + 8],  b[kk + 8],  c1);
                c2 = wmma_f32(a[kk + 16], b[kk + 16], c2);
                c3 = wmma_f32(a[kk + 24], b[kk + 24], c3);
            }
            v8f c = (c0 + c1) + (c2 + c3);

            // C layout: VGPR r holds row (r + 8*half), col = lm.
            if (diag_panel) {
                #pragma unroll
                for (int r = 0; r < 8; ++r) {
                    int grow = row_base + r + half * 8;
                    int gcol = n0 + sub * 16 + lm;
                    float v = c[r];
                    if (grow == gcol) { diag[r] = v; v -= BIGF; }
                    rmax[r] = fmaxf(rmax[r], v);
                }
            } else {
                #pragma unroll
                for (int r = 0; r < 8; ++r)
                    rmax[r] = fmaxf(rmax[r], c[r]);
            }
        }

        wait_asynccnt0();   // next panel landed (overlapped with compute)
        __syncthreads();
    }

    // Reduce across the 16 lanes of each half (masks 8..1 stay in-half)
    #pragma unroll
    for (int r = 0; r < 8; ++r) {
        #pragma unroll
        for (int m = 8; m >= 1; m >>= 1) {
            rmax[r] = fmaxf(rmax[r], __shfl_xor(rmax[r], m, 32));
            diag[r] += __shfl_xor(diag[r], m, 32);
        }
    }
    float partial = 0.f;
    #pragma unroll
    for (int r = 0; r < 8; ++r)
        partial += softplus_f(rmax[r] - diag[r]);   // 8 rows per half
    partial += __shfl_xor(partial, 16, 32);          // combine halves
    if (lane == 0) atomicAdd(&ws[1], partial);
}

// ---------------- Kernel 3: finalize ----------------
__global__ void finalize_kernel(const float* __restrict__ ws, float* __restrict__ out) {
    out[0] = 0.5f * (ws[0] + ws[1]) * (1.0f / (float)B_ROWS);
}

extern "C" void kernel_launch(void* const* d_in, const int* in_sizes, int n_in,
                              void* d_out, int out_size, void* d_ws, size_t ws_size,
                              hipStream_t stream) {
    const float* q    = (const float*)d_in[0];
    const float* dpos = (const float*)d_in[1];
    const float* dneg = (const float*)d_in[2];
    float* out = (float*)d_out;
    float* ws  = (float*)d_ws;

    hipMemsetAsync(ws, 0, 2 * sizeof(float), stream);
    pairwise_kernel<<<B_ROWS / 8, 256, 0, stream>>>(q, dpos, dneg, ws);
    gemm_rowmax_kernel<<<B_ROWS / BM, 128, 0, stream>>>(q, dpos, ws);
    finalize_kernel<<<1, 1, 0, stream>>>(ws, out);
}